// regionLayer_14302241095754
// MI455X (gfx1250) — compile-verified
//
#include <hip/hip_runtime.h>
#include <hip/hip_bf16.h>
#include <stdint.h>

// ---------------- problem constants ----------------
constexpr int kNB = 32, kNA = 5, kNC = 80, kNH = 76, kNW = 76;
constexpr int kMaxObj = 50;
constexpr int kCh = 5 + kNC;                 // 85
constexpr int kPlane = kNH * kNW;            // 5776
constexpr int kCellsPerB = kNA * kPlane;     // 28880
constexpr float kSil = 0.6f;

__constant__ float c_AW[kNA] = {18.3274f/32.f, 59.9827f/32.f, 106.8298f/32.f, 252.2502f/32.f, 312.6566f/32.f};
__constant__ float c_AH[kNA] = {21.6763f/32.f, 66.001f/32.f, 175.1789f/32.f, 112.889f/32.f, 293.385f/32.f};

// ---------------- CDNA5 helper types ----------------
typedef __attribute__((ext_vector_type(2))) float v2f;
typedef __attribute__((ext_vector_type(8))) float v8f;
typedef __attribute__((ext_vector_type(4))) unsigned int u32x4;
typedef __attribute__((ext_vector_type(8))) int i32x8;
typedef __attribute__((ext_vector_type(4))) int i32x4;

// fast reciprocal (v_rcp_f32) instead of the ~10-op precise div sequence
__device__ __forceinline__ float frcp(float x) {
#if __has_builtin(__builtin_amdgcn_rcpf)
  return __builtin_amdgcn_rcpf(x);
#else
  return 1.0f / x;
#endif
}
__device__ __forceinline__ float fsig(float x) { return frcp(1.0f + __expf(-x)); }

// Wave32 sum using V_WMMA_F32_16X16X4_F32 with B == ones.
// A: VGPR0=v, VGPR1=0  ->  A[m,0]=v(lane m), A[m,2]=v(lane m+16), A[m,1]=A[m,3]=0
// D[m,n] = v_m + v_{m+16} (replicated over n). Summing the 8 D VGPRs gives each
// half-wave the sum of its 8 rows; one shfl_xor(16) completes the 32-lane sum.
__device__ __forceinline__ float wave_sum(float v) {
#if __has_builtin(__builtin_amdgcn_wmma_f32_16x16x4_f32)
  v2f a; a[0] = v;    a[1] = 0.0f;
  v2f b; b[0] = 1.0f; b[1] = 1.0f;
  v8f c = {0.f,0.f,0.f,0.f,0.f,0.f,0.f,0.f};
  c = __builtin_amdgcn_wmma_f32_16x16x4_f32(false, a, false, b, (short)0, c, false, false);
  float s = c[0]+c[1]+c[2]+c[3]+c[4]+c[5]+c[6]+c[7];
  s += __shfl_xor(s, 16, 32);
  return s;
#else
  for (int off = 16; off > 0; off >>= 1) v += __shfl_xor(v, off, 32);
  return v;
#endif
}

// ---------------- kernel 0: zero the output scalar ----------------
__global__ void zero_kernel(float* out, int n) {
  int i = blockIdx.x * blockDim.x + threadIdx.x;
  if (i < n) out[i] = 0.0f;
}

// ---------------- kernel 1: per-batch GT preprocessing ----------------
// gtbox[b][o] = (gx,gy,gw,gh) in grid units, zeroed if invalid (zero boxes
// produce IoU==0 against any positive-area pred box, matching where(valid,...,0)).
__global__ __launch_bounds__(64) void prep_kernel(const float* __restrict__ target,
                                                  float4* __restrict__ gtbox) {
  int b = blockIdx.x, o = threadIdx.x;
  if (o >= kMaxObj) return;
  bool valid = true;
  for (int p = 0; p <= o; ++p)                       // valid = cumprod(tgt[:,:,1] > 0)
    valid = valid && (target[(size_t)(b * kMaxObj + p) * 5 + 1] > 0.0f);
  const float* t = target + (size_t)(b * kMaxObj + o) * 5;
  float4 g;
  g.x = valid ? t[1] * kNW : 0.0f;
  g.y = valid ? t[2] * kNH : 0.0f;
  g.z = valid ? t[3] * kNW : 0.0f;
  g.w = valid ? t[4] * kNH : 0.0f;
  gtbox[b * kMaxObj + o] = g;
}

// ---------------- kernel 2: global noobj-conf loss ----------------
// Per cell: pred box from x,y,w,h; silence test (any IoU vs GT > 0.6) done as
// inter > 0.6*union (no divisions); contribute 0.5*conf^2 when not silenced.
// GT boxes staged into LDS by the Tensor Data Mover (one DMA per block).
__global__ __launch_bounds__(256) void cell_kernel(const float* __restrict__ out,
                                                   const float* __restrict__ gtbox,
                                                   float* __restrict__ loss) {
  __shared__ float4 sGT[kMaxObj];
  __shared__ float  sWave[8];
  const int b   = blockIdx.y;
  const int tid = threadIdx.x;

  // ---- TDM: DMA 50*16B of GT boxes into LDS (wave 0 issues; D# per ISA §8) ----
  if (tid < 32) {
#if __has_builtin(__builtin_amdgcn_tensor_load_to_lds)
    const unsigned nElem = kMaxObj * 4;                       // 200 f32 = 800B tile
    unsigned long long gaddr = (unsigned long long)(const void*)(gtbox + (size_t)b * nElem);
    unsigned lds_off = (unsigned)(uintptr_t)(void*)&sGT[0];   // generic addr low 32b == LDS offset
    u32x4 g0; i32x8 g1; i32x4 g2 = (i32x4)0; i32x4 g3 = (i32x4)0;
    g0[0] = 1u;                                               // count=1, user descriptor
    g0[1] = lds_off;                                          // lds_addr
    g0[2] = (unsigned)gaddr;                                  // global_addr[31:0]
    g0[3] = ((unsigned)(gaddr >> 32) & 0x01FFFFFFu) | 0x80000000u; // addr[56:32] | type=2
    g1[0] = 0x00020000;                                       // data_size=4B
    g1[1] = (int)(nElem << 16);                               // tensor_dim0 low16
    g1[2] = (int)(1u << 16);                                  // tensor_dim0 hi | tensor_dim1=1
    g1[3] = (int)(nElem << 16);                               // tile_dim0
    g1[4] = 1;                                                // tile_dim1=1
    g1[5] = (int)nElem;                                       // tensor_dim0_stride
    g1[6] = (int)(nElem << 16);                               // tensor_dim1_stride low16
    g1[7] = 0;
    __builtin_amdgcn_tensor_load_to_lds(g0, g1, g2, g3, (i32x8)0, 0);  // 6-arg (clang-23) form
    __builtin_amdgcn_s_wait_tensorcnt(0);
#else
    for (int o = tid; o < kMaxObj; o += 32)
      sGT[o] = ((const float4*)gtbox)[(size_t)b * kMaxObj + o];
#endif
  }
  __syncthreads();

  const int idx = blockIdx.x * 256 + tid;
  float contrib = 0.0f;
  if (idx < kCellsPerB) {
    const int a = idx / kPlane;
    const int r = idx - a * kPlane;
    const int i = r % kNW;
    const int j = r / kNW;
    const float* base = out + ((size_t)(b * kNA + a) * kCh) * kPlane + r;
    const float xo = base[0];
    const float yo = base[kPlane];
    const float wo = base[2 * kPlane];
    const float ho = base[3 * kPlane];
    const float co = base[4 * kPlane];
    const float px = fsig(xo) + (float)i;
    const float py = fsig(yo) + (float)j;
    const float pw = __expf(wo) * c_AW[a];
    const float ph = __expf(ho) * c_AH[a];
    const float pxl = px - 0.5f * pw, pxh = px + 0.5f * pw;
    const float pyl = py - 0.5f * ph, pyh = py + 0.5f * ph;
    const float pa = pw * ph;
    bool sil = false;
#pragma unroll 2
    for (int o = 0; o < kMaxObj; ++o) {
      float4 g = sGT[o];                                 // ds_load_b128, broadcast
      float iw = fminf(pxh, g.x + 0.5f * g.z) - fmaxf(pxl, g.x - 0.5f * g.z);
      float ih = fminf(pyh, g.y + 0.5f * g.w) - fmaxf(pyl, g.y - 0.5f * g.w);
      iw = fmaxf(iw, 0.0f); ih = fmaxf(ih, 0.0f);
      float inter = iw * ih;
      float uni   = pa + g.z * g.w - inter;
      sil = sil || (inter > kSil * uni);                 // iou > 0.6 without dividing
    }
    if (!sil) { float conf = fsig(co); contrib = 0.5f * conf * conf; }
  }

  // full-EXEC WMMA wave reduction, then block reduction + one atomic
  float s = wave_sum(contrib);
  if ((tid & 31) == 0) sWave[tid >> 5] = s;
  __syncthreads();
  if (tid == 0) {
    float t = 0.0f;
    for (int k = 0; k < 8; ++k) t += sWave[k];
    atomicAdd(loss, t);
  }
}

// ---------------- kernel 3: scattered-cell corrections ----------------
// For each "owner" object (last valid object writing its flat cell, matching
// .at[].set last-wins): coord losses, class NLL, and conf-loss correction
// (add 5*(conf-best_iou)^2/2, remove the baseline 0.5*conf^2 if cell_kernel added it).
__global__ __launch_bounds__(64) void obj_kernel(const float* __restrict__ out,
                                                 const float* __restrict__ target,
                                                 const float* __restrict__ gtbox,
                                                 float* __restrict__ loss) {
  __shared__ int sFlat[kMaxObj];
  const int b = blockIdx.x, o = threadIdx.x;

  float gx = 0, gy = 0, gw = 0, gh = 0, tcls = 0;
  bool valid = false;
  int best_n = 0, gi = 0, gj = 0;
  if (o < kMaxObj) {
    valid = true;
    for (int p = 0; p <= o; ++p)
      valid = valid && (target[(size_t)(b * kMaxObj + p) * 5 + 1] > 0.0f);
    const float* t = target + (size_t)(b * kMaxObj + o) * 5;
    tcls = t[0];
    gx = t[1] * kNW; gy = t[2] * kNH; gw = t[3] * kNW; gh = t[4] * kNH;
    float best = -1.0f;
    for (int n = 0; n < kNA; ++n) {                      // best anchor by wh-IoU
      float inter = fminf(gw, c_AW[n]) * fminf(gh, c_AH[n]);
      float uni   = gw * gh + c_AW[n] * c_AH[n] - inter;
      float ratio = inter * frcp(uni);
      if (ratio > best) { best = ratio; best_n = n; }
    }
    gi = (int)gx; gj = (int)gy;
    gi = min(max(gi, 0), kNW - 1); gj = min(max(gj, 0), kNH - 1);
    sFlat[o] = valid ? (((b * kNA + best_n) * kNH + gj) * kNW + gi) : -1;
  }
  __syncthreads();

  bool owner = (o < kMaxObj) && valid;
  if (owner) {
    int my = sFlat[o];
    for (int p = o + 1; p < kMaxObj; ++p)
      if (sFlat[p] == my) { owner = false; break; }
  }
  if (!owner) return;

  const float aw = c_AW[best_n], ah = c_AH[best_n];
  const float* base = out + ((size_t)(b * kNA + best_n) * kCh) * kPlane + gj * kNW + gi;
  const float sigx = fsig(base[0]);
  const float sigy = fsig(base[kPlane]);
  const float wo   = base[2 * kPlane];
  const float ho   = base[3 * kPlane];
  const float conf = fsig(base[4 * kPlane]);
  const float pw = __expf(wo) * aw, ph = __expf(ho) * ah;
  const float px = sigx + (float)gi, py = sigy + (float)gj;

  // best_iou = IoU(gt, pred_at)
  float iw = fminf(px + 0.5f * pw, gx + 0.5f * gw) - fmaxf(px - 0.5f * pw, gx - 0.5f * gw);
  float ih = fminf(py + 0.5f * ph, gy + 0.5f * gh) - fmaxf(py - 0.5f * ph, gy - 0.5f * gh);
  iw = fmaxf(iw, 0.0f); ih = fmaxf(ih, 0.0f);
  float inter = iw * ih;
  float best_iou = inter * frcp(pw * ph + gw * gh - inter);

  // replicate cell_kernel's silence test at this cell to undo its baseline term
  const float4* gtb = (const float4*)gtbox + (size_t)b * kMaxObj;
  const float pxl = px - 0.5f * pw, pxh = px + 0.5f * pw;
  const float pyl = py - 0.5f * ph, pyh = py + 0.5f * ph;
  const float pa = pw * ph;
  bool sil = false;
  for (int q = 0; q < kMaxObj; ++q) {
    float4 g = gtb[q];
    float qiw = fminf(pxh, g.x + 0.5f * g.z) - fmaxf(pxl, g.x - 0.5f * g.z);
    float qih = fminf(pyh, g.y + 0.5f * g.w) - fmaxf(pyl, g.y - 0.5f * g.w);
    qiw = fmaxf(qiw, 0.0f); qih = fmaxf(qih, 0.0f);
    float qin = qiw * qih;
    sil = sil || (qin > kSil * (pa + g.z * g.w - qin));
  }

  const float tx = gx - (float)gi, ty = gy - (float)gj;
  const float tw = __logf(gw * frcp(aw)), th = __logf(gh * frcp(ah));
  float l = 0.5f * ((sigx - tx) * (sigx - tx) + (sigy - ty) * (sigy - ty) +
                    (wo - tw) * (wo - tw) + (ho - th) * (ho - th));
  l += 2.5f * (conf - best_iou) * (conf - best_iou);     // (sqrt(5)*d)^2 / 2
  if (!sil) l -= 0.5f * conf * conf;                     // remove cell_kernel baseline

  // class NLL at this cell (channels 5..84, stride kPlane)
  const float* cl = base + 5 * kPlane;
  int label = min(max((int)tcls, 0), kNC - 1);
  float m = -3.4e38f;
  for (int c = 0; c < kNC; ++c) m = fmaxf(m, cl[(size_t)c * kPlane]);
  float se = 0.0f;
  for (int c = 0; c < kNC; ++c) se += __expf(cl[(size_t)c * kPlane] - m);
  l += (m + __logf(se)) - cl[(size_t)label * kPlane];

  atomicAdd(loss, l);
}

// ---------------- host-side launch ----------------
extern "C" void kernel_launch(void* const* d_in, const int* in_sizes, int n_in,
                              void* d_out, int out_size, void* d_ws, size_t ws_size,
                              hipStream_t stream) {
  (void)in_sizes; (void)n_in; (void)ws_size;
  const float* d_output = (const float*)d_in[0];     // (32, 425, 76, 76) fp32
  const float* d_target = (const float*)d_in[1];     // (32, 250) fp32
  float* d_loss = (float*)d_out;                     // scalar fp32
  float* d_gtbox = (float*)d_ws;                     // 32*50*4 fp32 = 25.6 KB

  zero_kernel<<<(out_size + 255) / 256, 256, 0, stream>>>(d_loss, out_size);
  prep_kernel<<<kNB, 64, 0, stream>>>(d_target, (float4*)d_gtbox);
  dim3 grid((kCellsPerB + 255) / 256, kNB);
  cell_kernel<<<grid, 256, 0, stream>>>(d_output, d_gtbox, d_loss);
  obj_kernel<<<kNB, 64, 0, stream>>>(d_output, d_target, d_gtbox, d_loss);
}